// VanillaRNN_23931557773883
// MI455X (gfx1250) — compile-verified
//
#include <hip/hip_runtime.h>
#include <math.h>

typedef __attribute__((ext_vector_type(2))) float v2f;
typedef __attribute__((ext_vector_type(8))) float v8f;

#define HID 512
#define SEQ 256
#define NCLS 10

__global__ void rnn_zero_kernel(float* __restrict__ p, int n) {
    int i = blockIdx.x * blockDim.x + threadIdx.x;
    if (i < n) p[i] = 0.0f;
}

// One RNN time step:
//   hnext[n, m] = tanh( fc_hx[n]*x[m*SEQ + t] + (fc_hh @ hprev)[m, n] + b_h[m] )
// GEMM: M-dim = rows of fc_hh (m), N-dim = cols of hprev (n), K = 512.
// Each wave computes a 2x2 block of 16x16 fp32 WMMA tiles (32x32 output).
__global__ __launch_bounds__(256) void rnn_step_kernel(
    const float* __restrict__ fc_hh,
    const float* __restrict__ hprev,
    float* __restrict__ hnext,
    const float* __restrict__ x,
    const float* __restrict__ fc_hx,
    const float* __restrict__ b_h,
    int t)
{
    const int wave = threadIdx.x >> 5;
    const int lane = threadIdx.x & 31;
    const int half = lane >> 4;      // 0: lanes 0-15, 1: lanes 16-31
    const int l    = lane & 15;

    const int W    = blockIdx.x * 8 + wave;   // 256 waves total
    const int row0 = (W >> 4) << 5;           // m-tile base (0..480 step 32)
    const int col0 = (W & 15) << 5;           // n-tile base

    v8f c00 = {}, c01 = {}, c10 = {}, c11 = {};

    const float* Arow0 = fc_hh + (size_t)(row0 + l) * HID;        // ti=0
    const float* Arow1 = fc_hh + (size_t)(row0 + 16 + l) * HID;   // ti=1

    for (int k = 0; k < HID; k += 4) {
        const int kk = k + 2 * half;  // K pair handled by this lane half
        // A fragments: lane holds A[m, kk], A[m, kk+1] (contiguous)
        v2f a0 = *(const v2f*)(Arow0 + kk);
        v2f a1 = *(const v2f*)(Arow1 + kk);
        // B fragments: VGPR0 = B[kk, n], VGPR1 = B[kk+1, n]
        v2f b0, b1;
        b0.x = hprev[(size_t)kk * HID + col0 + l];
        b0.y = hprev[(size_t)(kk + 1) * HID + col0 + l];
        b1.x = hprev[(size_t)kk * HID + col0 + 16 + l];
        b1.y = hprev[(size_t)(kk + 1) * HID + col0 + 16 + l];

        c00 = __builtin_amdgcn_wmma_f32_16x16x4_f32(false, a0, false, b0, (short)0, c00, false, false);
        c01 = __builtin_amdgcn_wmma_f32_16x16x4_f32(false, a0, false, b1, (short)0, c01, false, false);
        c10 = __builtin_amdgcn_wmma_f32_16x16x4_f32(false, a1, false, b0, (short)0, c10, false, false);
        c11 = __builtin_amdgcn_wmma_f32_16x16x4_f32(false, a1, false, b1, (short)0, c11, false, false);
    }

    // Epilogue with transpose: C/D layout -> VGPR v, lanes half: M = v + 8*half, N = l
    const int n0 = col0 + l;
    const int n1 = col0 + 16 + l;
    const float fx0 = fc_hx[n0];
    const float fx1 = fc_hx[n1];

#pragma unroll
    for (int v = 0; v < 8; ++v) {
        const int m0 = row0 + 8 * half + v;       // ti=0 rows
        const int m1 = m0 + 16;                   // ti=1 rows
        const float xm0 = x[(size_t)m0 * SEQ + t];
        const float xm1 = x[(size_t)m1 * SEQ + t];
        const float bh0 = b_h[m0];
        const float bh1 = b_h[m1];

        hnext[(size_t)n0 * HID + m0] = tanhf(fx0 * xm0 + c00[v] + bh0);
        hnext[(size_t)n0 * HID + m1] = tanhf(fx0 * xm1 + c10[v] + bh1);
        hnext[(size_t)n1 * HID + m0] = tanhf(fx1 * xm0 + c01[v] + bh0);
        hnext[(size_t)n1 * HID + m1] = tanhf(fx1 * xm1 + c11[v] + bh1);
    }
}

// Head: p[b,c] = sum_k h[k,b] * fc_ph[c,k] + b_p[c]; softmax over c.
// One wave per batch row b.
__global__ __launch_bounds__(256) void rnn_head_kernel(
    const float* __restrict__ h,
    const float* __restrict__ fc_ph,
    const float* __restrict__ b_p,
    float* __restrict__ out)
{
    const int wave = threadIdx.x >> 5;
    const int lane = threadIdx.x & 31;
    const int b = blockIdx.x * 8 + wave;   // 64 blocks * 8 waves = 512 rows

    float acc[NCLS];
#pragma unroll
    for (int c = 0; c < NCLS; ++c) acc[c] = 0.0f;

    for (int k = lane; k < HID; k += 32) {
        const float hv = h[(size_t)k * HID + b];
#pragma unroll
        for (int c = 0; c < NCLS; ++c) acc[c] += hv * fc_ph[(size_t)c * HID + k];
    }

#pragma unroll
    for (int off = 16; off > 0; off >>= 1) {
#pragma unroll
        for (int c = 0; c < NCLS; ++c) acc[c] += __shfl_xor(acc[c], off, 32);
    }

    if (lane == 0) {
        float p[NCLS];
        float mx = -3.4e38f;
#pragma unroll
        for (int c = 0; c < NCLS; ++c) { p[c] = acc[c] + b_p[c]; mx = fmaxf(mx, p[c]); }
        float s = 0.0f;
#pragma unroll
        for (int c = 0; c < NCLS; ++c) { p[c] = __expf(p[c] - mx); s += p[c]; }
        const float inv = 1.0f / s;
#pragma unroll
        for (int c = 0; c < NCLS; ++c) out[(size_t)b * NCLS + c] = p[c] * inv;
    }
}

extern "C" void kernel_launch(void* const* d_in, const int* in_sizes, int n_in,
                              void* d_out, int out_size, void* d_ws, size_t ws_size,
                              hipStream_t stream) {
    const float* x     = (const float*)d_in[0];   // [512, 256]
    const float* fc_hx = (const float*)d_in[1];   // [512, 1]
    const float* fc_hh = (const float*)d_in[2];   // [512, 512]
    const float* fc_ph = (const float*)d_in[3];   // [10, 512]
    const float* b_h   = (const float*)d_in[4];   // [512]
    const float* b_p   = (const float*)d_in[5];   // [10]
    float* out = (float*)d_out;                   // [512, 10]

    float* h0 = (float*)d_ws;                     // double-buffered hidden state
    float* h1 = h0 + (size_t)HID * HID;

    rnn_zero_kernel<<<(HID * HID + 255) / 256, 256, 0, stream>>>(h0, HID * HID);

    for (int t = 0; t < SEQ; ++t) {
        const float* src = (t & 1) ? h1 : h0;
        float*       dst = (t & 1) ? h0 : h1;
        rnn_step_kernel<<<32, 256, 0, stream>>>(fc_hh, src, dst, x, fc_hx, b_h, t);
    }
    // SEQ even -> final state is in h0
    rnn_head_kernel<<<64, 256, 0, stream>>>(h0, fc_ph, b_p, out);
}